// SegNet_24592982737597
// MI455X (gfx1250) — compile-verified
//
#include <hip/hip_runtime.h>
#include <math.h>

typedef __attribute__((ext_vector_type(2))) float v2f;
typedef __attribute__((ext_vector_type(8))) float v8f;

#define SPBLK 128   // superpixel blocks
#define EPS 1e-5f

static inline int cdiv_h(int a, int b) { return (a + b - 1) / b; }

// =======================================================================
// GEMM via V_WMMA_F32_16X16X4_F32:  out[d,p] = sum_c wt[c*D+d] * in[c*N+p]
// One wave owns a 16-pixel column strip across the FULL D = DT*16 output
// channels: the B fragment (bandwidth-critical [C,N] stream) is loaded once
// per k-step and reused by DT WMMAs.  N multiple of 16, K multiple of 4.
// f32 A 16x4 layout: lanes 0-15 -> {K=0,K=1} in v0,v1; lanes 16-31 -> {K=2,K=3}.
// f32 B 4x16 symmetric. C/D: VGPR r -> M=r (lanes 0-15) / M=8+r (lanes 16-31).
// =======================================================================
template <int DT>
__global__ void gemm_wmma_strip(const float* __restrict__ in, const float* __restrict__ wt,
                                float* __restrict__ out, int N, int K) {
  const int D = DT * 16;
  int tiles = N >> 4;
  int wave = (int)((blockIdx.x * blockDim.x + threadIdx.x) >> 5);
  if (wave >= tiles) return;                       // wave-uniform exit
  int lane = threadIdx.x & 31;
  int half = lane >> 4;                            // 0: lanes 0-15, 1: lanes 16-31
  int l = lane & 15;
  int p0 = wave << 4;

  const float* bptr = in + (size_t)(half << 1) * N + p0 + l;  // B[k=half*2][p0+l]
  const float* aptr = wt + (size_t)(half << 1) * D + l;       // A base (+dt*16 per tile)
  const size_t bstep = (size_t)4 * N;
  const size_t astep = (size_t)4 * D;

  v8f acc[DT];
#pragma unroll
  for (int dt = 0; dt < DT; ++dt) acc[dt] = (v8f){0.f, 0.f, 0.f, 0.f, 0.f, 0.f, 0.f, 0.f};

#if __has_builtin(__builtin_amdgcn_wmma_f32_16x16x4_f32)
  for (int k0 = 0; k0 < K; k0 += 4) {
    v2f b;
    b.x = bptr[0];
    b.y = bptr[N];
    __builtin_prefetch(bptr + 2 * bstep, 0, 0);    // speculative: safe if OOB
#pragma unroll
    for (int dt = 0; dt < DT; ++dt) {
      v2f a;
      a.x = aptr[dt * 16];
      a.y = aptr[D + dt * 16];
      acc[dt] = __builtin_amdgcn_wmma_f32_16x16x4_f32(false, a, false, b, (short)0,
                                                      acc[dt], false, false);
    }
    bptr += bstep;
    aptr += astep;
  }
#else
  for (int k0 = 0; k0 < K; ++k0) {
    float bv = in[(size_t)k0 * N + p0 + l];
#pragma unroll
    for (int dt = 0; dt < DT; ++dt)
      for (int r = 0; r < 8; ++r)
        acc[dt][r] = fmaf(wt[(size_t)k0 * D + dt * 16 + r + (half << 3)], bv, acc[dt][r]);
  }
#endif
#pragma unroll
  for (int dt = 0; dt < DT; ++dt)
    for (int r = 0; r < 8; ++r)
      out[(size_t)(dt * 16 + r + (half << 3)) * N + p0 + l] = acc[dt][r];
}

// x [N, C] (pixel-major) -> out [C, N] (channel-major)
__global__ void k_transpose(const float* __restrict__ x, float* __restrict__ out,
                            int N, int C) {
  int p = blockIdx.x * blockDim.x + threadIdx.x;
  if (p >= N) return;
  for (int c = 0; c < C; ++c) out[(size_t)c * N + p] = x[(size_t)p * C + c];
}

// nearest-downsample index (256x256 source) to h x w + histogram counts
__global__ void k_idx_down(const int* __restrict__ index, int* __restrict__ out,
                           float* __restrict__ counts, int h, int w) {
  int t = blockIdx.x * blockDim.x + threadIdx.x;
  if (t >= h * w) return;
  int i = t / w, j = t - i * w;
  int ri = (i * 256) / h, ci = (j * 256) / w;
  int bb = index[ri * 256 + ci];
  out[t] = bb;
  atomicAdd(&counts[bb], 1.0f);
}

// segment sums: sums[b*C+c] += xw[c,p] for idx[p]==b, LDS-staged
__global__ void k_seg_sum(const float* __restrict__ xw, const int* __restrict__ idx,
                          float* __restrict__ sums, int N, int C, int chunk) {
  extern __shared__ float lsum[];                  // SPBLK * C floats
  int total = SPBLK * C;
  for (int i = threadIdx.x; i < total; i += blockDim.x) lsum[i] = 0.f;
  __syncthreads();
  int base = blockIdx.x * chunk;
  int work = C * chunk;
  for (int lin = threadIdx.x; lin < work; lin += blockDim.x) {
    int c  = lin / chunk;
    int pl = lin - c * chunk;
    int p  = base + pl;
    if (p < N) atomicAdd(&lsum[idx[p] * C + c], xw[(size_t)c * N + p]);
  }
  __syncthreads();
  for (int i = threadIdx.x; i < total; i += blockDim.x) {
    float v = lsum[i];
    if (v != 0.f) atomicAdd(&sums[i], v);
  }
}

__global__ void k_seg_mean(const float* __restrict__ sums, const float* __restrict__ counts,
                           float* __restrict__ means, int C) {
  int i = blockIdx.x * blockDim.x + threadIdx.x;
  if (i >= SPBLK * C) return;
  float cnt = counts[i / C];
  means[i] = sums[i] / (cnt == 0.f ? 1.f : cnt);
}

// t[b,o] = sum_c means[b,c] * Wm[c,o]   (128 x C x C, tiny)
__global__ void k_small_gemm(const float* __restrict__ means, const float* __restrict__ Wm,
                             float* __restrict__ t, int C) {
  int i = blockIdx.x * blockDim.x + threadIdx.x;
  if (i >= SPBLK * C) return;
  int b = i / C, o = i - b * C;
  float s = 0.f;
  for (int c = 0; c < C; ++c) s = fmaf(means[b * C + c], Wm[c * C + o], s);
  t[i] = s;
}

// agg[p,c] = sum_{q!=p} mask[p,q] * exp(-||t_q - t_p||^2) * means[q,c]
__global__ void k_agg(const float* __restrict__ t, const float* __restrict__ means,
                      const float* __restrict__ mask, float* __restrict__ agg, int C) {
  __shared__ float wq[SPBLK];
  int p = blockIdx.x, q = threadIdx.x;
  float quad = 0.f;
  for (int o = 0; o < C; ++o) {
    float d = t[q * C + o] - t[p * C + o];
    quad = fmaf(d, d, quad);
  }
  wq[q] = (q == p) ? 0.f : expf(-quad) * mask[p * SPBLK + q];
  __syncthreads();
  for (int c = q; c < C; c += SPBLK) {
    float s = 0.f;
    for (int qq = 0; qq < SPBLK; ++qq) s = fmaf(wq[qq], means[qq * C + c], s);
    agg[p * C + c] = s;
  }
}

// feat[c,p] = xw[c,p] + agg[idx[p],c] (in place) + per-channel sum/sumsq stats
__global__ void k_feat_stats(float* __restrict__ xw, const float* __restrict__ agg,
                             const int* __restrict__ idx, float* __restrict__ stats,
                             int N, int C) {
  int c = blockIdx.y;
  int p = blockIdx.x * blockDim.x + threadIdx.x;
  float v = 0.f;
  if (p < N) {
    v = xw[(size_t)c * N + p] + agg[idx[p] * C + c];
    xw[(size_t)c * N + p] = v;
  }
  __shared__ float ssum[256], ssq[256];
  ssum[threadIdx.x] = (p < N) ? v : 0.f;
  ssq[threadIdx.x]  = (p < N) ? v * v : 0.f;
  __syncthreads();
  for (int s = blockDim.x >> 1; s > 0; s >>= 1) {
    if ((int)threadIdx.x < s) {
      ssum[threadIdx.x] += ssum[threadIdx.x + s];
      ssq[threadIdx.x]  += ssq[threadIdx.x + s];
    }
    __syncthreads();
  }
  if (threadIdx.x == 0) {
    atomicAdd(&stats[c], ssum[0]);
    atomicAdd(&stats[C + c], ssq[0]);
  }
}

// per-channel BN coefficients: coef[c]=scale, coef[C+c]=shift
__global__ void k_bn_coef(const float* __restrict__ stats, const float* __restrict__ gamma,
                          const float* __restrict__ beta, float* __restrict__ coef,
                          int C, float invN) {
  int c = blockIdx.x * blockDim.x + threadIdx.x;
  if (c >= C) return;
  float mu   = stats[c] * invN;
  float var  = stats[C + c] * invN - mu * mu;
  float rstd = rsqrtf(var + EPS);
  float sc   = gamma[c] * rstd;
  coef[c]     = sc;
  coef[C + c] = beta[c] - mu * sc;
}

__global__ void k_bn_apply(const float* __restrict__ feat, const float* __restrict__ coef,
                           float* __restrict__ out, int N, int C) {
  int t = blockIdx.x * blockDim.x + threadIdx.x;
  if (t >= N * C) return;
  int c = t / N;
  out[t] = fmaf(feat[t], coef[c], coef[C + c]);
}

// MaxPool2d(kernel=3, stride=2, padding=1): [C,h,w] -> [C,h/2,w/2]
__global__ void k_maxpool(const float* __restrict__ in, float* __restrict__ out,
                          int C, int h, int w) {
  int ho = h >> 1, wo = w >> 1;
  int t = blockIdx.x * blockDim.x + threadIdx.x;
  if (t >= C * ho * wo) return;
  int c = t / (ho * wo);
  int r = (t / wo) % ho;
  int col = t % wo;
  float m = -INFINITY;
  for (int di = -1; di <= 1; ++di) {
    int i = 2 * r + di;
    if (i < 0 || i >= h) continue;
    for (int dj = -1; dj <= 1; ++dj) {
      int j = 2 * col + dj;
      if (j < 0 || j >= w) continue;
      m = fmaxf(m, in[((size_t)c * h + i) * w + j]);
    }
  }
  out[t] = m;
}

// bilinear align_corners=True upsample [C,h,w] -> [C,256,256]
__global__ void k_upsample(const float* __restrict__ in, float* __restrict__ out,
                           int C, int h, int w) {
  int t = blockIdx.x * blockDim.x + threadIdx.x;
  if (t >= C * 65536) return;
  int c = t >> 16;
  int i = (t >> 8) & 255;
  int j = t & 255;
  float py = i * ((h - 1) / 255.0f);
  float px = j * ((w - 1) / 255.0f);
  int y0 = (int)py; if (y0 > h - 1) y0 = h - 1;
  int x0 = (int)px; if (x0 > w - 1) x0 = w - 1;
  int y1 = y0 + 1 < h ? y0 + 1 : h - 1;
  int x1 = x0 + 1 < w ? x0 + 1 : w - 1;
  float fy = py - y0, fx = px - x0;
  const float* base = in + (size_t)c * h * w;
  float v00 = base[y0 * w + x0], v01 = base[y0 * w + x1];
  float v10 = base[y1 * w + x0], v11 = base[y1 * w + x1];
  float top = v00 * (1.f - fx) + v01 * fx;
  float bot = v10 * (1.f - fx) + v11 * fx;
  out[t] = top * (1.f - fy) + bot * fy;
}

// =======================================================================
// Host-side: one GCN level (all launches on `stream`; capture-safe only)
// =======================================================================
static void run_gcn(const float* in_cm, int N, int K, int D,
                    const float* wt, const float* wm, const float* gamma, const float* beta,
                    const float* mask, const int* idx, const float* counts,
                    float* A, float* out_norm,
                    float* sums, float* means, float* tb, float* agg,
                    float* stats, float* coef, hipStream_t s) {
  hipMemsetAsync(sums, 0, SPBLK * D * sizeof(float), s);
  hipMemsetAsync(stats, 0, 2 * D * sizeof(float), s);
  int waves = N / 16;
  if (D == 64)
    gemm_wmma_strip<4><<<cdiv_h(waves * 32, 256), 256, 0, s>>>(in_cm, wt, A, N, K);
  else
    gemm_wmma_strip<1><<<cdiv_h(waves * 32, 256), 256, 0, s>>>(in_cm, wt, A, N, K);
  int chunk = 1024;
  k_seg_sum<<<cdiv_h(N, chunk), 256, SPBLK * D * sizeof(float), s>>>(A, idx, sums, N, D, chunk);
  k_seg_mean<<<cdiv_h(SPBLK * D, 256), 256, 0, s>>>(sums, counts, means, D);
  k_small_gemm<<<cdiv_h(SPBLK * D, 256), 256, 0, s>>>(means, wm, tb, D);
  k_agg<<<SPBLK, SPBLK, 0, s>>>(tb, means, mask, agg, D);
  dim3 fsg((unsigned)cdiv_h(N, 256), (unsigned)D);
  k_feat_stats<<<fsg, 256, 0, s>>>(A, agg, idx, stats, N, D);
  k_bn_coef<<<1, 256, 0, s>>>(stats, gamma, beta, coef, D, 1.0f / (float)N);
  k_bn_apply<<<cdiv_h(N * D, 256), 256, 0, s>>>(A, coef, out_norm, N, D);
}

extern "C" void kernel_launch(void* const* d_in, const int* in_sizes, int n_in,
                              void* d_out_v, int out_size, void* d_ws, size_t ws_size,
                              hipStream_t stream) {
  (void)in_sizes; (void)n_in; (void)out_size; (void)ws_size;
  const float* x     = (const float*)d_in[0];   // [256,256,64] pixel-major
  const int*   index = (const int*)d_in[1];     // [256,256]
  const float* mask  = (const float*)d_in[2];   // [128,128]
  const float* W[4]  = {(const float*)d_in[3],  (const float*)d_in[7],
                        (const float*)d_in[11], (const float*)d_in[15]};
  const float* M[4]  = {(const float*)d_in[4],  (const float*)d_in[8],
                        (const float*)d_in[12], (const float*)d_in[16]};
  const float* G[4]  = {(const float*)d_in[5],  (const float*)d_in[9],
                        (const float*)d_in[13], (const float*)d_in[17]};
  const float* Bt[4] = {(const float*)d_in[6],  (const float*)d_in[10],
                        (const float*)d_in[14], (const float*)d_in[18]};
  const float* wall = (const float*)d_in[19];
  const float* mall = (const float*)d_in[20];
  const float* gall = (const float*)d_in[21];
  const float* ball = (const float*)d_in[22];

  float* dout = (float*)d_out_v;
  float* ws   = (float*)d_ws;

  // workspace layout (floats)
  float* A    = ws;                     // 64*65536 : xw / feat (in place)
  float* Bf   = ws + 4194304;           // 64*65536 : normalized features
  float* Cb   = ws + 8388608;           // 64*16384 : pooled features
  float* misc = ws + 9437184;
  float* sums   = misc;                 // 128*64
  float* means  = misc + 8192;          // 128*64
  float* tb     = misc + 16384;         // 128*64
  float* agg    = misc + 24576;         // 128*64
  float* stats  = misc + 32768;         // 2*64
  float* coef   = misc + 32896;         // 2*64
  float* counts = misc + 33024;         // 4*128 (res 256/128/64/32)
  int*   idxb   = (int*)(misc + 33536);
  int* idx256 = idxb;                   // 65536
  int* idx128 = idxb + 65536;           // 16384
  int* idx64  = idxb + 81920;           // 4096
  int* idx32  = idxb + 86016;           // 1024

  // d_out layout: finall[16*65536] | f1u | f2u | f3u | f4u (each 64*65536)
  float* finall = dout;
  float* f1u = dout + 1048576;
  float* f2u = dout + 5242880;
  float* f3u = dout + 9437184;
  float* f4u = dout + 13631488;

  // index pyramids + superpixel counts per resolution
  hipMemsetAsync(counts, 0, 512 * sizeof(float), stream);
  k_idx_down<<<cdiv_h(65536, 256), 256, 0, stream>>>(index, idx256, counts + 0,   256, 256);
  k_idx_down<<<cdiv_h(16384, 256), 256, 0, stream>>>(index, idx128, counts + 128, 128, 128);
  k_idx_down<<<cdiv_h(4096, 256),  256, 0, stream>>>(index, idx64,  counts + 256, 64, 64);
  k_idx_down<<<cdiv_h(1024, 256),  256, 0, stream>>>(index, idx32,  counts + 384, 32, 32);

  // ---- stage 1 (256x256) ----
  k_transpose<<<cdiv_h(65536, 256), 256, 0, stream>>>(x, Bf, 65536, 64);
  run_gcn(Bf, 65536, 64, 64, W[0], M[0], G[0], Bt[0], mask, idx256, counts + 0,
          A, Bf, sums, means, tb, agg, stats, coef, stream);
  k_maxpool<<<cdiv_h(64 * 128 * 128, 256), 256, 0, stream>>>(Bf, Cb, 64, 256, 256);
  k_upsample<<<cdiv_h(64 * 65536, 256), 256, 0, stream>>>(Cb, f1u, 64, 128, 128);

  // ---- stage 2 (128x128) ----
  run_gcn(Cb, 16384, 64, 64, W[1], M[1], G[1], Bt[1], mask, idx128, counts + 128,
          A, Bf, sums, means, tb, agg, stats, coef, stream);
  k_maxpool<<<cdiv_h(64 * 64 * 64, 256), 256, 0, stream>>>(Bf, Cb, 64, 128, 128);
  k_upsample<<<cdiv_h(64 * 65536, 256), 256, 0, stream>>>(Cb, f2u, 64, 64, 64);

  // ---- stage 3 (64x64) ----
  run_gcn(Cb, 4096, 64, 64, W[2], M[2], G[2], Bt[2], mask, idx64, counts + 256,
          A, Bf, sums, means, tb, agg, stats, coef, stream);
  k_maxpool<<<cdiv_h(64 * 32 * 32, 256), 256, 0, stream>>>(Bf, Cb, 64, 64, 64);
  k_upsample<<<cdiv_h(64 * 65536, 256), 256, 0, stream>>>(Cb, f3u, 64, 32, 32);

  // ---- stage 4 (32x32) ----
  run_gcn(Cb, 1024, 64, 64, W[3], M[3], G[3], Bt[3], mask, idx32, counts + 384,
          A, Bf, sums, means, tb, agg, stats, coef, stream);
  k_maxpool<<<cdiv_h(64 * 16 * 16, 256), 256, 0, stream>>>(Bf, Cb, 64, 32, 32);
  k_upsample<<<cdiv_h(64 * 65536, 256), 256, 0, stream>>>(Cb, f4u, 64, 16, 16);

  // ---- final stage: concat(f1u..f4u) = [256, 65536] in d_out, 256 -> 16 ----
  run_gcn(f1u, 65536, 256, 16, wall, mall, gall, ball, mask, idx256, counts + 0,
          A, finall, sums, means, tb, agg, stats, coef, stream);
}